// ShallowEncoder_12472585028189
// MI455X (gfx1250) — compile-verified
//
#include <hip/hip_runtime.h>

typedef __attribute__((ext_vector_type(2))) float v2f;
typedef __attribute__((ext_vector_type(8))) float v8f;

#define C_DIM 64

// ---- degree init: deg = 1.0 (self loop) ----
__global__ void k_deg_init(float* __restrict__ deg, int n) {
  int i = blockIdx.x * blockDim.x + threadIdx.x;
  if (i < n) deg[i] = 1.0f;
}

// ---- degree accumulate over edge destinations ----
__global__ void k_deg_accum(const int* __restrict__ dst, float* __restrict__ deg, int e) {
  int i = blockIdx.x * blockDim.x + threadIdx.x;
  if (i < e) atomicAdd(&deg[dst[i]], 1.0f);
}

// ---- deg -> deg^{-1/2} in place ----
__global__ void k_rsqrt(float* __restrict__ deg, int n) {
  int i = blockIdx.x * blockDim.x + threadIdx.x;
  if (i < n) {
    float d = deg[i];
    deg[i] = (d > 0.0f) ? rsqrtf(d) : 0.0f;
  }
}

// ---- h = x @ W via V_WMMA_F32_16X16X4_F32 (one wave per 16-row tile) ----
// B operands for each k-step are pre-swizzled in LDS into per-lane order:
//   sB[kstep][nb][lane][2]  (v2f per lane -> single ds_load_b64, bank-conflict-free)
__global__ void k_gemm_wmma(const float* __restrict__ x, const float* __restrict__ W,
                            float* __restrict__ h, int n_nodes) {
  __shared__ __align__(16) float sB[16 * 4 * 32 * 2];  // 16 KB

  // Stage W permuted: element i = ((kstep*4 + nb)*32 + L)*2 + p holds
  //   W[kstep*4 + 2*(L>>4) + p][nb*16 + (L&15)]
  for (int i = threadIdx.x; i < 16 * 4 * 32 * 2; i += blockDim.x) {
    int p     = i & 1;
    int L     = (i >> 1) & 31;
    int nb    = (i >> 6) & 3;
    int kstep = i >> 8;
    int row   = kstep * 4 + ((L >> 4) << 1) + p;
    int col   = nb * 16 + (L & 15);
    sB[i] = W[row * C_DIM + col];
  }
  __syncthreads();

  const int lane = threadIdx.x & 31;
  const int wave = threadIdx.x >> 5;
  const int tile = blockIdx.x * (blockDim.x >> 5) + wave;
  const int m0   = tile * 16;
  if (m0 >= n_nodes) return;           // wave-uniform: EXEC stays all-ones

  const int half = lane >> 4;          // 0: lanes 0-15, 1: lanes 16-31
  const int l    = lane & 15;
  const int koff = half * 2;           // A-operand K offset for upper half-wave
  int arow = m0 + l;
  if (arow >= n_nodes) arow = n_nodes - 1;   // clamp (keeps EXEC full)
  const float* xrow = x + (size_t)arow * C_DIM;

  const v2f* bv = reinterpret_cast<const v2f*>(sB);

  v8f c0 = {}, c1 = {}, c2 = {}, c3 = {};

#pragma unroll
  for (int kstep = 0; kstep < 16; ++kstep) {
    // A 16x4 fp32: lane<16 holds K=kk,kk+1 of row m0+l; lane>=16 holds K=kk+2,kk+3
    v2f a = *reinterpret_cast<const v2f*>(xrow + kstep * 4 + koff);
    // B 4x16 fp32 tiles, pre-packed per lane: one b64 load each, nb at +256B steps
    const v2f* bk = bv + kstep * 128 + lane;
    v2f b0 = bk[0];
    v2f b1 = bk[32];
    v2f b2 = bk[64];
    v2f b3 = bk[96];
    c0 = __builtin_amdgcn_wmma_f32_16x16x4_f32(false, a, false, b0, (short)0, c0, false, false);
    c1 = __builtin_amdgcn_wmma_f32_16x16x4_f32(false, a, false, b1, (short)0, c1, false, false);
    c2 = __builtin_amdgcn_wmma_f32_16x16x4_f32(false, a, false, b2, (short)0, c2, false, false);
    c3 = __builtin_amdgcn_wmma_f32_16x16x4_f32(false, a, false, b3, (short)0, c3, false, false);
  }

  // C/D layout: VGPR v -> row m0+v+8*half, col l (+16 per tile)
#pragma unroll
  for (int v = 0; v < 8; ++v) {
    int row = m0 + v + half * 8;
    if (row < n_nodes) {
      float* hr = h + (size_t)row * C_DIM;
      hr[l]      = c0[v];
      hr[l + 16] = c1[v];
      hr[l + 32] = c2[v];
      hr[l + 48] = c3[v];
    }
  }
}

// ---- out = b + dinv^2 * h  (self-loop term + bias; also initializes out) ----
__global__ void k_self_bias(const float* __restrict__ h, const float* __restrict__ dinv,
                            const float* __restrict__ b, float* __restrict__ out, int total) {
  int i = blockIdx.x * blockDim.x + threadIdx.x;
  if (i < total) {
    int node = i >> 6, c = i & 63;
    float di = dinv[node];
    out[i] = b[c] + di * di * h[i];
  }
}

// ---- edge scatter: out[dst] += dinv[src]*dinv[dst] * h[src]; 16 threads/edge, float4 ----
__global__ void k_scatter(const int* __restrict__ src, const int* __restrict__ dst,
                          const float* __restrict__ dinv, const float* __restrict__ h,
                          float* __restrict__ out, int n_edges) {
  int gid = blockIdx.x * blockDim.x + threadIdx.x;
  int e = gid >> 4;
  if (e >= n_edges) return;
  int q = gid & 15;
  int s = src[e], d = dst[e];
  float norm = dinv[s] * dinv[d];
  float4 m = reinterpret_cast<const float4*>(h + (size_t)s * C_DIM)[q];
  float* op = out + (size_t)d * C_DIM + q * 4;
  atomicAdd(op + 0, norm * m.x);
  atomicAdd(op + 1, norm * m.y);
  atomicAdd(op + 2, norm * m.z);
  atomicAdd(op + 3, norm * m.w);
}

extern "C" void kernel_launch(void* const* d_in, const int* in_sizes, int n_in,
                              void* d_out, int out_size, void* d_ws, size_t ws_size,
                              hipStream_t stream) {
  const float* x  = (const float*)d_in[0];
  const int*   ei = (const int*)d_in[1];     // JAX default x64-off: int32, shape [2, E] flat
  const float* W  = (const float*)d_in[2];
  const float* b  = (const float*)d_in[3];
  float* out = (float*)d_out;

  const int n_nodes = in_sizes[0] / C_DIM;
  const int n_edges = in_sizes[1] / 2;
  const int* src = ei;
  const int* dst = ei + n_edges;

  // workspace: [deg/dinv : N floats][h : N*64 floats]
  float* deg = (float*)d_ws;
  float* h   = deg + n_nodes;

  const int B = 256;
  k_deg_init <<<(n_nodes + B - 1) / B, B, 0, stream>>>(deg, n_nodes);
  k_deg_accum<<<(n_edges + B - 1) / B, B, 0, stream>>>(dst, deg, n_edges);
  k_rsqrt    <<<(n_nodes + B - 1) / B, B, 0, stream>>>(deg, n_nodes);

  const int tiles = (n_nodes + 15) / 16;          // one wave32 per tile, 8 waves/block
  k_gemm_wmma<<<(tiles + 7) / 8, B, 0, stream>>>(x, W, h, n_nodes);

  const int total = n_nodes * C_DIM;
  k_self_bias<<<(total + B - 1) / B, B, 0, stream>>>(h, deg, b, out, total);

  const long long work = (long long)n_edges * 16;
  k_scatter<<<(int)((work + B - 1) / B), B, 0, stream>>>(src, dst, deg, h, out, n_edges);
}